// MultiHeadPointTransformer_86827058856422
// MI455X (gfx1250) — compile-verified
//
#include <hip/hip_runtime.h>
#include <math.h>

// ---------------------------------------------------------------------------
// MultiHeadPointTransformer on gfx1250 (MI455X)
//   N=100000 nodes (divisible by 16), E=800000 edges, 64 channels (4 heads x 16)
//   fp32 throughout; GEMMs use V_WMMA_F32_16X16X4_F32.
// ---------------------------------------------------------------------------

typedef __attribute__((ext_vector_type(2))) float v2f;
typedef __attribute__((ext_vector_type(8))) float v8f;

#define IN_C  64
#define OUT_C 64
#define HD    16

// float atomic-max via signed/unsigned integer atomics (valid for all floats
// when the destination is initialized to -inf).
__device__ __forceinline__ void atomicMaxF(float* addr, float val) {
  if (val >= 0.0f) atomicMax((int*)addr, __float_as_int(val));
  else             atomicMin((unsigned int*)addr, __float_as_uint(val));
}

// ---------------------------------------------------------------------------
// K0: init amax = -inf, den = 0, num = 0
// ---------------------------------------------------------------------------
__global__ void init_kernel(float* __restrict__ amax, float* __restrict__ den,
                            float* __restrict__ num, int n) {
  int i = blockIdx.x * blockDim.x + threadIdx.x;
  int stride = gridDim.x * blockDim.x;
  for (; i < n; i += stride) {
    amax[i] = -INFINITY;
    den[i]  = 0.0f;
    num[i]  = 0.0f;
  }
}

// ---------------------------------------------------------------------------
// K1: fused node projections  v / a_src / a_dst = x @ W_{lin,src,dst}
//     One wave = one 16x16 output tile; K-loop of 16 WMMA f32 16x16x4 steps.
//     W layout: [HEADS][64][16] flat -> head-j column tile is contiguous
//     [64][16] at offset j*1024.  Outputs stored as [N][64], col = h*16+c'.
// ---------------------------------------------------------------------------
__global__ void proj_kernel(const float* __restrict__ x,
                            const float* __restrict__ Wl,
                            const float* __restrict__ Ws,
                            const float* __restrict__ Wd,
                            float* __restrict__ v_out,
                            float* __restrict__ as_out,
                            float* __restrict__ ad_out,
                            int nTiles) {
  int waveId = blockIdx.x * (blockDim.x >> 5) + (threadIdx.x >> 5);
  if (waveId >= nTiles * 12) return;           // uniform per-wave: EXEC stays full
  int lane = threadIdx.x & 31;

  int tile = waveId / 12;
  int sub  = waveId % 12;
  int mat  = sub >> 2;                         // 0=v, 1=a_src, 2=a_dst
  int j    = sub & 3;                          // head / 16-col tile

  const float* W = (mat == 0) ? Wl : (mat == 1 ? Ws : Wd);
  float*       O = (mat == 0) ? v_out : (mat == 1 ? as_out : ad_out);
  const float* Wj = W + (size_t)j * (IN_C * HD);   // [64][16]

  int m     = lane & 15;                       // A: M-row, B/C/D: N-col
  int khalf = (lane >> 4) * 2;                 // lanes 16-31 hold K+2,K+3

  const float* Arow = x + (size_t)(tile * 16 + m) * IN_C;

  v8f c;
  #pragma unroll
  for (int r = 0; r < 8; ++r) c[r] = 0.0f;

  #pragma unroll
  for (int kk = 0; kk < 16; ++kk) {
    int kb = kk * 4 + khalf;
    v2f a, b;
    a.x = Arow[kb];
    a.y = Arow[kb + 1];
    b.x = Wj[(size_t)kb * HD + m];
    b.y = Wj[(size_t)(kb + 1) * HD + m];
    c = __builtin_amdgcn_wmma_f32_16x16x4_f32(false, a, false, b,
                                              (short)0, c, false, false);
  }

  int rowBase = tile * 16 + ((lane >> 4) * 8);
  int col     = j * 16 + m;
  #pragma unroll
  for (int r = 0; r < 8; ++r)
    O[(size_t)(rowBase + r) * OUT_C + col] = c[r];
}

// ---------------------------------------------------------------------------
// Edge kernels: one wave per edge, 2 channels per lane (c = 2*lane, 2*lane+1).
// delta_c = (pos_d - pos_s) . W_pos[h][:,c'] + b_pos[c]
// alpha_c = a_dst[d,c] - a_src[s,c] + delta_c
// ---------------------------------------------------------------------------
__global__ void edge_max_kernel(const int* __restrict__ ei,
                                const float* __restrict__ pos,
                                const float* __restrict__ a_src,
                                const float* __restrict__ a_dst,
                                const float* __restrict__ Wp,  // [H][3][16]
                                const float* __restrict__ bp,  // [64]
                                float* __restrict__ amax,
                                int E, int Et) {
  int e = blockIdx.x * (blockDim.x >> 5) + (threadIdx.x >> 5);
  if (e >= Et) return;
  int lane = threadIdx.x & 31;

  int s, d;
  if (e < E) { s = ei[e]; d = ei[E + e]; }
  else       { s = d = e - E; }                // self-loop

  float rx = pos[d * 3 + 0] - pos[s * 3 + 0];
  float ry = pos[d * 3 + 1] - pos[s * 3 + 1];
  float rz = pos[d * 3 + 2] - pos[s * 3 + 2];

  float2 adv = ((const float2*)(a_dst + (size_t)d * OUT_C))[lane];
  float2 asv = ((const float2*)(a_src + (size_t)s * OUT_C))[lane];

  int c0 = lane * 2;
  #pragma unroll
  for (int t = 0; t < 2; ++t) {
    int c  = c0 + t;
    int h  = c >> 4, cp = c & 15;
    float delta = rx * Wp[h * 48 + cp] + ry * Wp[h * 48 + 16 + cp] +
                  rz * Wp[h * 48 + 32 + cp] + bp[c];
    float ad = t ? adv.y : adv.x;
    float as = t ? asv.y : asv.x;
    atomicMaxF(&amax[(size_t)d * OUT_C + c], ad - as + delta);
  }
}

// Fused pass: den += ex ; num += ex * (v[src] + delta).  (attn normalization is
// applied once per node later: out = num/den == sum(ex*m)/sum(ex).)
__global__ void edge_acc_kernel(const int* __restrict__ ei,
                                const float* __restrict__ pos,
                                const float* __restrict__ a_src,
                                const float* __restrict__ a_dst,
                                const float* __restrict__ vbuf,
                                const float* __restrict__ Wp,
                                const float* __restrict__ bp,
                                const float* __restrict__ amax,
                                float* __restrict__ den,
                                float* __restrict__ num,
                                int E, int Et) {
  int e = blockIdx.x * (blockDim.x >> 5) + (threadIdx.x >> 5);
  if (e >= Et) return;
  int lane = threadIdx.x & 31;

  int s, d;
  if (e < E) { s = ei[e]; d = ei[E + e]; }
  else       { s = d = e - E; }

  float rx = pos[d * 3 + 0] - pos[s * 3 + 0];
  float ry = pos[d * 3 + 1] - pos[s * 3 + 1];
  float rz = pos[d * 3 + 2] - pos[s * 3 + 2];

  float2 adv = ((const float2*)(a_dst + (size_t)d * OUT_C))[lane];
  float2 asv = ((const float2*)(a_src + (size_t)s * OUT_C))[lane];
  float2 vv  = ((const float2*)(vbuf  + (size_t)s * OUT_C))[lane];
  float2 mx  = ((const float2*)(amax  + (size_t)d * OUT_C))[lane];

  int c0 = lane * 2;
  #pragma unroll
  for (int t = 0; t < 2; ++t) {
    int c  = c0 + t;
    int h  = c >> 4, cp = c & 15;
    float delta = rx * Wp[h * 48 + cp] + ry * Wp[h * 48 + 16 + cp] +
                  rz * Wp[h * 48 + 32 + cp] + bp[c];
    float ad = t ? adv.y : adv.x;
    float as = t ? asv.y : asv.x;
    float vj = t ? vv.y  : vv.x;
    float m  = t ? mx.y  : mx.x;
    float ex = __expf((ad - as + delta) - m);
    size_t idx = (size_t)d * OUT_C + c;
    atomicAdd(&den[idx], ex);
    atomicAdd(&num[idx], ex * (vj + delta));
  }
}

// ---------------------------------------------------------------------------
// K4: projection MLP via WMMA.  NORM: A = num/max(den,1e-16); RELU on output.
//     Accumulator is initialized with the bias (broadcast per column).
// ---------------------------------------------------------------------------
template <bool NORM, bool RELU>
__global__ void mlp_kernel(const float* __restrict__ Anum,
                           const float* __restrict__ Aden,
                           const float* __restrict__ W,     // [64][64]
                           const float* __restrict__ bias,  // [64]
                           float* __restrict__ out,
                           int nTiles) {
  int waveId = blockIdx.x * (blockDim.x >> 5) + (threadIdx.x >> 5);
  if (waveId >= nTiles * 4) return;
  int lane = threadIdx.x & 31;

  int tile = waveId >> 2;
  int j    = waveId & 3;
  int m     = lane & 15;
  int khalf = (lane >> 4) * 2;

  const float* An = Anum + (size_t)(tile * 16 + m) * OUT_C;
  const float* Ad = NORM ? (Aden + (size_t)(tile * 16 + m) * OUT_C) : nullptr;

  float bval = bias[j * 16 + m];
  v8f c;
  #pragma unroll
  for (int r = 0; r < 8; ++r) c[r] = bval;

  #pragma unroll
  for (int kk = 0; kk < 16; ++kk) {
    int kb = kk * 4 + khalf;
    v2f a, b;
    if (NORM) {
      a.x = An[kb]     / fmaxf(Ad[kb],     1e-16f);
      a.y = An[kb + 1] / fmaxf(Ad[kb + 1], 1e-16f);
    } else {
      a.x = An[kb];
      a.y = An[kb + 1];
    }
    b.x = W[(size_t)kb * OUT_C + j * 16 + m];
    b.y = W[(size_t)(kb + 1) * OUT_C + j * 16 + m];
    c = __builtin_amdgcn_wmma_f32_16x16x4_f32(false, a, false, b,
                                              (short)0, c, false, false);
  }

  int rowBase = tile * 16 + ((lane >> 4) * 8);
  int col     = j * 16 + m;
  #pragma unroll
  for (int r = 0; r < 8; ++r) {
    float val = c[r];
    if (RELU) val = fmaxf(val, 0.0f);
    out[(size_t)(rowBase + r) * OUT_C + col] = val;
  }
}

// ---------------------------------------------------------------------------
extern "C" void kernel_launch(void* const* d_in, const int* in_sizes, int n_in,
                              void* d_out, int out_size, void* d_ws, size_t ws_size,
                              hipStream_t stream) {
  const float* x   = (const float*)d_in[0];
  const float* pos = (const float*)d_in[1];
  const int*   ei  = (const int*)  d_in[2];
  const float* Wl  = (const float*)d_in[3];
  const float* Ws  = (const float*)d_in[4];
  const float* Wd  = (const float*)d_in[5];
  const float* Wp  = (const float*)d_in[6];
  const float* bp  = (const float*)d_in[7];
  const float* W1  = (const float*)d_in[8];
  const float* b1  = (const float*)d_in[9];
  const float* W2  = (const float*)d_in[10];
  const float* b2  = (const float*)d_in[11];
  float* out = (float*)d_out;

  const int N  = in_sizes[0] / IN_C;
  const int E  = in_sizes[2] / 2;
  const int Et = E + N;                 // + self-loops
  const size_t nc = (size_t)N * OUT_C;

  // workspace layout (fp32): v | a_src | a_dst | amax | den | num ; h1 reuses v
  float* ws_f    = (float*)d_ws;
  float* v_buf   = ws_f;
  float* as_buf  = ws_f + 1 * nc;
  float* ad_buf  = ws_f + 2 * nc;
  float* amax    = ws_f + 3 * nc;
  float* den     = ws_f + 4 * nc;
  float* num     = ws_f + 5 * nc;
  float* h1      = v_buf;               // v dead after edge_acc

  const int nTiles = N / 16;            // N = 100000 -> 6250 exact tiles

  // K0: init accumulators
  init_kernel<<<2048, 256, 0, stream>>>(amax, den, num, (int)nc);

  // K1: fused projections (WMMA f32)
  {
    int waves = nTiles * 12;
    proj_kernel<<<(waves + 7) / 8, 256, 0, stream>>>(x, Wl, Ws, Wd,
                                                     v_buf, as_buf, ad_buf, nTiles);
  }

  // K2 + K3: edge passes (one wave per edge)
  {
    int blocks = (Et + 7) / 8;
    edge_max_kernel<<<blocks, 256, 0, stream>>>(ei, pos, as_buf, ad_buf,
                                                Wp, bp, amax, E, Et);
    edge_acc_kernel<<<blocks, 256, 0, stream>>>(ei, pos, as_buf, ad_buf, v_buf,
                                                Wp, bp, amax, den, num, E, Et);
  }

  // K4: MLP (WMMA f32): h1 = relu((num/den)@W1 + b1); out = h1@W2 + b2
  {
    int waves = nTiles * 4;
    mlp_kernel<true, true><<<(waves + 7) / 8, 256, 0, stream>>>(
        num, den, W1, b1, h1, nTiles);
    mlp_kernel<false, false><<<(waves + 7) / 8, 256, 0, stream>>>(
        h1, nullptr, W2, b2, out, nTiles);
  }
}